// GCN_42528766165301
// MI455X (gfx1250) — compile-verified
//
#include <hip/hip_runtime.h>

typedef __attribute__((ext_vector_type(2))) float v2f;
typedef __attribute__((ext_vector_type(8))) float v8f;
typedef __attribute__((ext_vector_type(4))) int v4i;

#define DIV_UP(a, b) (((a) + (b) - 1) / (b))

// gfx1250 async global->LDS copy path (guarded: falls back to plain LDS stores)
#if defined(__gfx1250__) && __has_builtin(__builtin_amdgcn_global_load_async_to_lds_b128) && \
    __has_builtin(__builtin_amdgcn_s_wait_asynccnt)
#define GCN_ASYNC_LDS 1
typedef __attribute__((address_space(1))) v4i* gcn_gptr_v4i;
typedef __attribute__((address_space(3))) v4i* gcn_lptr_v4i;
#endif

// ---------------------------------------------------------------------------
// Elementwise / graph-structure kernels
// ---------------------------------------------------------------------------
__global__ void gcn_fill_kernel(float* __restrict__ p, float v, int n) {
  int i = blockIdx.x * blockDim.x + threadIdx.x;
  if (i < n) p[i] = v;
}

__global__ void gcn_deg_count_kernel(const long long* __restrict__ dst,
                                     float* __restrict__ deg, int E) {
  int e = blockIdx.x * blockDim.x + threadIdx.x;
  if (e < E) atomicAdd(&deg[(int)dst[e]], 1.0f);
}

__global__ void gcn_dinv_kernel(float* __restrict__ deg, int n) {
  int i = blockIdx.x * blockDim.x + threadIdx.x;
  if (i < n) deg[i] = rsqrtf(deg[i]);  // deg >= 1 due to self-loop
}

__global__ void gcn_edge_norm_kernel(const long long* __restrict__ src,
                                     const long long* __restrict__ dst,
                                     const float* __restrict__ dinv,
                                     float* __restrict__ norm, int E) {
  int e = blockIdx.x * blockDim.x + threadIdx.x;
  if (e < E) norm[e] = dinv[(int)src[e]] * dinv[(int)dst[e]];
}

// acc[i, :] = bias + (dinv[i]^2) * hW[i, :]   (self-loop contribution + bias)
__global__ void gcn_self_init_kernel(const float* __restrict__ hW,
                                     const float* __restrict__ dinv,
                                     const float* __restrict__ bias,
                                     float* __restrict__ acc, int total) {
  int idx = blockIdx.x * blockDim.x + threadIdx.x;
  if (idx < total) {
    int i = idx >> 7;
    int c = idx & 127;
    float di = dinv[i];
    acc[idx] = bias[c] + di * di * hW[idx];
  }
}

// One thread per (edge, 4-feature group): acc[dst] += norm[e] * hW[src]
// hW (51 MB) is resident in the 192 MB L2, so both the random gather and the
// f32 atomics are L2-local.
__global__ void gcn_edge_scatter_kernel(const long long* __restrict__ src,
                                        const long long* __restrict__ dst,
                                        const float* __restrict__ norm,
                                        const float* __restrict__ hW,
                                        float* __restrict__ acc, int E) {
  int idx = blockIdx.x * blockDim.x + threadIdx.x;
  if (idx < E * 32) {
    int e = idx >> 5;
    int c = (idx & 31) << 2;
    int s = (int)src[e];
    int d = (int)dst[e];
    float nm = norm[e];
    const float4 v = *(const float4*)(hW + (size_t)s * 128 + c);
    float* p = acc + (size_t)d * 128 + c;
    atomicAdd(p + 0, v.x * nm);
    atomicAdd(p + 1, v.y * nm);
    atomicAdd(p + 2, v.z * nm);
    atomicAdd(p + 3, v.w * nm);
  }
}

__global__ void gcn_relu_kernel(float* __restrict__ p, int n) {
  int i = blockIdx.x * blockDim.x + threadIdx.x;
  if (i < n) p[i] = fmaxf(p[i], 0.0f);
}

// ---------------------------------------------------------------------------
// Pooling kernels
// ---------------------------------------------------------------------------
__global__ void gcn_count_kernel(const long long* __restrict__ batch,
                                 float* __restrict__ counts, int n) {
  int i = blockIdx.x * blockDim.x + threadIdx.x;
  if (i < n) atomicAdd(&counts[(int)batch[i]], 1.0f);
}

__global__ void gcn_pool_sum_kernel(const float* __restrict__ h,
                                    const long long* __restrict__ batch,
                                    float* __restrict__ pooled, int n) {
  int idx = blockIdx.x * blockDim.x + threadIdx.x;
  if (idx < n * 32) {
    int i = idx >> 5;
    int c = (idx & 31) << 2;
    int g = (int)batch[i];
    const float4 v = *(const float4*)(h + (size_t)i * 128 + c);
    float* p = pooled + (size_t)g * 128 + c;
    atomicAdd(p + 0, v.x);
    atomicAdd(p + 1, v.y);
    atomicAdd(p + 2, v.z);
    atomicAdd(p + 3, v.w);
  }
}

__global__ void gcn_pool_div_kernel(float* __restrict__ pooled,
                                    const float* __restrict__ counts, int total) {
  int idx = blockIdx.x * blockDim.x + threadIdx.x;
  if (idx < total) pooled[idx] /= fmaxf(counts[idx >> 7], 1.0f);
}

// ---------------------------------------------------------------------------
// WMMA f32 GEMM: C[M,N] = A[M,K] @ W[K,N] (+bias, +relu)
//  - W staged in LDS once per block via async global->LDS DMA (pad +8 dwords:
//    2*LDN % 64 == 16, so lane halves hit disjoint bank groups ->
//    conflict-free ds_load).
//  - Each wave owns MT 16-row strips; all NT = N/16 output tiles live in
//    registers, so each A element is loaded once and each B fragment feeds
//    MT WMMAs (load/WMMA ratio ~= (2*MT + 2*NT) / (MT*NT)).
//  - Requires K % 4 == 0, N % 32 == 0; M handled by wave-uniform exit.
// fuse: 0 = none, 1 = bias+relu
// ---------------------------------------------------------------------------
template <int NT, int MT>
__global__ void __launch_bounds__(256)
gcn_wmma_gemm_f32_kernel(const float* __restrict__ A, const float* __restrict__ W,
                         const float* __restrict__ bias, float* __restrict__ C,
                         int M, int N, int K, int fuse) {
  extern __shared__ float lds_w[];
  const int LDN = N + 8;

  // Cooperative W -> LDS stage (all threads participate, before any exit).
  const int nvec = (K * N) >> 2;  // float4 chunks
  for (int i4 = threadIdx.x; i4 < nvec; i4 += blockDim.x) {
    int k = i4 / (N >> 2);
    int n = (i4 - k * (N >> 2)) << 2;
#ifdef GCN_ASYNC_LDS
    __builtin_amdgcn_global_load_async_to_lds_b128(
        (gcn_gptr_v4i)(v4i*)(W + (size_t)k * N + n),
        (gcn_lptr_v4i)(v4i*)(lds_w + k * LDN + n),
        /*offset=*/0, /*cpol=*/0);
#else
    *(float4*)(lds_w + k * LDN + n) = *(const float4*)(W + (size_t)k * N + n);
#endif
  }
#ifdef GCN_ASYNC_LDS
  __builtin_amdgcn_s_wait_asynccnt(0);
#endif
  __syncthreads();

  const int wave = threadIdx.x >> 5;
  const int lane = threadIdx.x & 31;
  const int strip = blockIdx.x * (blockDim.x >> 5) + wave;
  const int m0 = strip * 16 * MT;
  if (m0 >= M) return;  // wave-uniform: EXEC stays all-ones for WMMA below

  const int half = lane >> 4;  // 0: K pair (0,1) / rows M=r ; 1: K pair (2,3) / rows M=r+8
  const int l = lane & 15;

  v8f acc[MT][NT] = {};

  // A fragment source rows (16x4 f32 layout: both lane halves cover M=0..15,
  // halves differ in the K sub-pair).
  const float* arow[MT];
#pragma unroll
  for (int mt = 0; mt < MT; ++mt)
    arow[mt] = A + (size_t)(m0 + mt * 16 + l) * K;

  for (int k0 = 0; k0 < K; k0 += 4) {
    const int kh = k0 + 2 * half;
    v2f a[MT];
#pragma unroll
    for (int mt = 0; mt < MT; ++mt) {
      a[mt].x = arow[mt][kh + 0];
      a[mt].y = arow[mt][kh + 1];
    }
#pragma unroll
    for (int t = 0; t < NT; ++t) {
      const int n0 = t * 16;
      // B 4x16 layout: VGPR0 holds rows K=k0 (lanes 0-15) / k0+2 (lanes
      // 16-31); VGPR1 rows k0+1 / k0+3.  Served from LDS, bank-conflict-free.
      v2f b;
      b.x = lds_w[(kh + 0) * LDN + n0 + l];
      b.y = lds_w[(kh + 1) * LDN + n0 + l];
#pragma unroll
      for (int mt = 0; mt < MT; ++mt) {
        acc[mt][t] = __builtin_amdgcn_wmma_f32_16x16x4_f32(
            /*neg_a=*/false, a[mt], /*neg_b=*/false, b,
            /*c_mod=*/(short)0, acc[mt][t], /*reuse_a=*/false, /*reuse_b=*/false);
      }
    }
  }

  // C/D layout: VGPR r -> row base+r (lanes 0-15) and row base+r+8 (lanes 16-31)
#pragma unroll
  for (int mt = 0; mt < MT; ++mt) {
#pragma unroll
    for (int t = 0; t < NT; ++t) {
      const int n0 = t * 16;
#pragma unroll
      for (int r = 0; r < 8; ++r) {
        int row = m0 + mt * 16 + r + half * 8;
        int col = n0 + l;
        float v = acc[mt][t][r];
        if (fuse) {
          v += bias[col];
          v = fmaxf(v, 0.0f);
        }
        C[(size_t)row * N + col] = v;
      }
    }
  }
}

// Final FC: 32 -> 1
__global__ void gcn_fc_final_kernel(const float* __restrict__ gin,
                                    const float* __restrict__ w,
                                    const float* __restrict__ b,
                                    float* __restrict__ out, int G, int K) {
  int g = blockIdx.x * blockDim.x + threadIdx.x;
  if (g < G) {
    float s = b[0];
    for (int k = 0; k < K; ++k) s += gin[(size_t)g * K + k] * w[k];
    out[g] = s;
  }
}

// ---------------------------------------------------------------------------
// Host driver
// ---------------------------------------------------------------------------
extern "C" void kernel_launch(void* const* d_in, const int* in_sizes, int n_in,
                              void* d_out, int out_size, void* d_ws, size_t ws_size,
                              hipStream_t stream) {
  const float* x = (const float*)d_in[0];
  const long long* edge_index = (const long long*)d_in[1];
  const long long* batch = (const long long*)d_in[2];
  const int E = in_sizes[1] / 2;
  const int N = in_sizes[0] / 128;  // 100000
  const int G = out_size;           // 64 graphs
  const long long* src = edge_index;
  const long long* dst = edge_index + E;

  const float* cw[3] = {(const float*)d_in[3], (const float*)d_in[5], (const float*)d_in[7]};
  const float* cb[3] = {(const float*)d_in[4], (const float*)d_in[6], (const float*)d_in[8]};
  const float* fw[6];
  const float* fb[6];
  for (int i = 0; i < 6; ++i) {
    fw[i] = (const float*)d_in[9 + 2 * i];
    fb[i] = (const float*)d_in[10 + 2 * i];
  }

  // Workspace carve-up (256B aligned)
  char* ws = (char*)d_ws;
  size_t off = 0;
  auto alloc = [&](size_t bytes) -> float* {
    float* p = (float*)(ws + off);
    off += (bytes + 255) & ~(size_t)255;
    return p;
  };
  float* dinv   = alloc((size_t)N * 4);           // deg, then rsqrt(deg)
  float* norm   = alloc((size_t)E * 4);
  float* hW     = alloc((size_t)N * 128 * 4);
  float* acc    = alloc((size_t)N * 128 * 4);
  float* pooled = alloc((size_t)G * 128 * 4);
  float* counts = alloc((size_t)G * 4);
  float* g1     = alloc((size_t)G * 256 * 4);
  float* g2     = alloc((size_t)G * 256 * 4);
  (void)ws_size; (void)n_in;

  const int T = 256;
  auto lds_bytes = [](int K, int N) -> size_t { return (size_t)K * (N + 8) * 4; };

  // Degrees (self-loop gives each node deg 1 to start) and symmetric norms
  gcn_fill_kernel<<<DIV_UP(N, T), T, 0, stream>>>(dinv, 1.0f, N);
  gcn_deg_count_kernel<<<DIV_UP(E, T), T, 0, stream>>>(dst, dinv, E);
  gcn_dinv_kernel<<<DIV_UP(N, T), T, 0, stream>>>(dinv, N);
  gcn_edge_norm_kernel<<<DIV_UP(E, T), T, 0, stream>>>(src, dst, dinv, norm, E);

  // 3 GCN conv layers: WMMA GEMM (MT=2 strips/wave), then edge aggregation.
  const float* hin = x;
  for (int layer = 0; layer < 3; ++layer) {
    const int strips = DIV_UP(N, 32);  // 3125, exact (32 rows per wave)
    gcn_wmma_gemm_f32_kernel<8, 2><<<DIV_UP(strips, 8), 256, lds_bytes(128, 128), stream>>>(
        hin, cw[layer], nullptr, hW, N, 128, 128, /*fuse=*/0);
    gcn_self_init_kernel<<<DIV_UP(N * 128, T), T, 0, stream>>>(hW, dinv, cb[layer], acc, N * 128);
    gcn_edge_scatter_kernel<<<DIV_UP(E * 32, T), T, 0, stream>>>(src, dst, norm, hW, acc, E);
    gcn_relu_kernel<<<DIV_UP(N * 128, T), T, 0, stream>>>(acc, N * 128);
    hin = acc;
  }

  // Global mean pool
  gcn_fill_kernel<<<1, T, 0, stream>>>(counts, 0.0f, G);
  gcn_fill_kernel<<<DIV_UP(G * 128, T), T, 0, stream>>>(pooled, 0.0f, G * 128);
  gcn_count_kernel<<<DIV_UP(N, T), T, 0, stream>>>(batch, counts, N);
  gcn_pool_sum_kernel<<<DIV_UP(N * 32, T), T, 0, stream>>>(acc, batch, pooled, N);
  gcn_pool_div_kernel<<<DIV_UP(G * 128, T), T, 0, stream>>>(pooled, counts, G * 128);

  // MLP head: 128 -> 128 -> 256 -> 128 -> 64 -> 32 -> 1 (WMMA except last)
  gcn_wmma_gemm_f32_kernel<8, 1><<<1, 128, lds_bytes(128, 128), stream>>>(pooled, fw[0], fb[0], g1, G, 128, 128, 1);
  gcn_wmma_gemm_f32_kernel<16, 1><<<1, 128, lds_bytes(128, 256), stream>>>(g1, fw[1], fb[1], g2, G, 256, 128, 1);
  gcn_wmma_gemm_f32_kernel<8, 1><<<1, 128, lds_bytes(256, 128), stream>>>(g2, fw[2], fb[2], g1, G, 128, 256, 1);
  gcn_wmma_gemm_f32_kernel<4, 1><<<1, 128, lds_bytes(128, 64), stream>>>(g1, fw[3], fb[3], g2, G, 64, 128, 1);
  gcn_wmma_gemm_f32_kernel<2, 1><<<1, 128, lds_bytes(64, 32), stream>>>(g2, fw[4], fb[4], g1, G, 32, 64, 1);
  gcn_fc_final_kernel<<<1, 64, 0, stream>>>(g1, fw[5], fb[5], (float*)d_out, G, 32);
}